// SGCNConv_76647986365162
// MI455X (gfx1250) — compile-verified
//
#include <hip/hip_runtime.h>
#include <math.h>

#define DIN 128
#define DOUT 128

typedef __attribute__((ext_vector_type(2))) float v2f;
typedef __attribute__((ext_vector_type(8))) float v8f;

__device__ __forceinline__ float sigmoidf_(float z) {
    return 1.0f / (1.0f + __expf(-z));
}

// ---------------------------------------------------------------------------
// K1: Y = x @ W for three weight matrices, via V_WMMA_F32_16X16X4_F32.
// grid.x = N/16 row tiles (N is a multiple of 16 -> no partial tiles),
// grid.z = {loop, in, out}; block = 256 (8 waves), wave w computes the
// 16x16 output tile at columns [16w, 16w+16).
// A-frag layout (16x4 f32): lanes 0-15 -> M=lane, VGPR0=K, VGPR1=K+1;
//                           lanes 16-31 -> M=lane-16, VGPR0=K+2, VGPR1=K+3.
// B-frag layout (4x16 f32): lanes 0-15 -> N=lane, K rows {K, K+1};
//                           lanes 16-31 -> N=lane-16, K rows {K+2, K+3}.
// ---------------------------------------------------------------------------
__global__ __launch_bounds__(256)
void sgcn_gemm_kernel(const float* __restrict__ x,
                      const float* __restrict__ W0,
                      const float* __restrict__ W1,
                      const float* __restrict__ W2,
                      float* __restrict__ Y0,
                      float* __restrict__ Y1,
                      float* __restrict__ Y2,
                      int N)
{
    const int wave = threadIdx.x >> 5;
    const int lane = threadIdx.x & 31;
    const int hv   = lane >> 4;        // 0 for lanes 0-15, 1 for lanes 16-31
    const int l16  = lane & 15;
    const int m0   = blockIdx.x * 16;
    const int n0   = wave * 16;        // 8 waves cover all 128 output columns

    const float* W;
    float*       Y;
    if (blockIdx.z == 0)      { W = W0; Y = Y0; }
    else if (blockIdx.z == 1) { W = W1; Y = Y1; }
    else                      { W = W2; Y = Y2; }

    const int row = m0 + l16;          // always < N (N % 16 == 0)
    const float* xA = x + (size_t)row * DIN + 2 * hv;
    const float* wB = W + (size_t)(2 * hv) * DOUT + n0 + l16;

    v8f acc = {};
    #pragma unroll 4
    for (int k = 0; k < DIN; k += 4) {
        v2f a = *(const v2f*)(xA + k);
        v2f b;
        b.x = wB[(size_t)k * DOUT];
        b.y = wB[(size_t)(k + 1) * DOUT];
        acc = __builtin_amdgcn_wmma_f32_16x16x4_f32(
            false, a, false, b, (short)0, acc, false, false);
    }

    // D layout: VGPR j -> row m0 + 8*hv + j, col n0 + l16.
    // All rows valid (N % 16 == 0) -> unconditional store clause.
    float* yb = Y + (size_t)(m0 + 8 * hv) * DOUT + n0 + l16;
    #pragma unroll
    for (int j = 0; j < 8; ++j) {
        yb[(size_t)j * DOUT] = acc[j];
    }
}

// ---------------------------------------------------------------------------
// K2: one wave per node. Computes the three 128->1 gate dots with a wave
// reduction; applies sigmoid(g_loop)*(Y_loop + b_loop) in-place on d_out;
// stores raw g_in / g_out dots (label bias + sigmoid applied per edge).
// ---------------------------------------------------------------------------
__global__ __launch_bounds__(256)
void sgcn_gate_kernel(const float* __restrict__ x,
                      const float* __restrict__ Wg_loop,
                      const float* __restrict__ bg_loop,   // [1]
                      const float* __restrict__ Wg_in,
                      const float* __restrict__ Wg_out,
                      const float* __restrict__ b_loop,    // [128]
                      float* __restrict__ out,             // holds Y_loop raw
                      float* __restrict__ g_in,
                      float* __restrict__ g_out,
                      int N)
{
    const int node = (int)((blockIdx.x * blockDim.x + threadIdx.x) >> 5);
    const int lane = threadIdx.x & 31;
    if (node >= N) return;

    const float4 xv = *(const float4*)(x + (size_t)node * DIN + lane * 4);
    const float4 wl = *(const float4*)(Wg_loop + lane * 4);
    const float4 wi = *(const float4*)(Wg_in  + lane * 4);
    const float4 wo = *(const float4*)(Wg_out + lane * 4);

    float dl = xv.x * wl.x + xv.y * wl.y + xv.z * wl.z + xv.w * wl.w;
    float di = xv.x * wi.x + xv.y * wi.y + xv.z * wi.z + xv.w * wi.w;
    float dd = xv.x * wo.x + xv.y * wo.y + xv.z * wo.z + xv.w * wo.w;

    #pragma unroll
    for (int off = 16; off > 0; off >>= 1) {
        dl += __shfl_xor(dl, off, 32);
        di += __shfl_xor(di, off, 32);
        dd += __shfl_xor(dd, off, 32);
    }

    const float gl = sigmoidf_(dl + bg_loop[0]);
    float4 yl = *(float4*)(out + (size_t)node * DIN + lane * 4);
    const float4 bl = *(const float4*)(b_loop + lane * 4);
    yl.x = gl * (yl.x + bl.x);
    yl.y = gl * (yl.y + bl.y);
    yl.z = gl * (yl.z + bl.z);
    yl.w = gl * (yl.w + bl.w);
    *(float4*)(out + (size_t)node * DIN + lane * 4) = yl;

    if (lane == 0) { g_in[node] = di; g_out[node] = dd; }
}

// ---------------------------------------------------------------------------
// K3: one wave per edge (grid-stride). Each lane handles 4 contiguous output
// channels (float4 gathers, per-float hardware f32 atomics for the scatter).
// Yi/Yo/out (3 x 51.2 MB) fit in the 192 MB L2 -> L2-resident scatter phase.
// ---------------------------------------------------------------------------
__global__ __launch_bounds__(256)
void sgcn_edge_kernel(const float* __restrict__ Yi,
                      const float* __restrict__ Yo,
                      const float* __restrict__ g_in,
                      const float* __restrict__ g_out,
                      const int*   __restrict__ edge_index,  // [2,E]
                      const int*   __restrict__ edge_label,  // [E]
                      const float* __restrict__ b_in,        // [L,128]
                      const float* __restrict__ b_in_g,      // [L]
                      const float* __restrict__ b_out,       // [L,128]
                      const float* __restrict__ b_out_g,     // [L]
                      float* __restrict__ out,
                      int E)
{
    const int lane   = threadIdx.x & 31;
    const int wave   = (int)((blockIdx.x * blockDim.x + threadIdx.x) >> 5);
    const int nwaves = (int)((gridDim.x * blockDim.x) >> 5);
    const int c      = lane * 4;

    for (int e = wave; e < E; e += nwaves) {
        const int s = edge_index[e];
        const int d = edge_index[E + e];
        const int l = edge_label[e];

        // "in" direction: gather src, scatter to dst
        {
            const float gate = sigmoidf_(g_in[s] + b_in_g[l]);
            const float4 m = *(const float4*)(Yi + (size_t)s * DOUT + c);
            const float4 b = *(const float4*)(b_in + (size_t)l * DOUT + c);
            float* o = out + (size_t)d * DOUT + c;
            unsafeAtomicAdd(o + 0, gate * (m.x + b.x));
            unsafeAtomicAdd(o + 1, gate * (m.y + b.y));
            unsafeAtomicAdd(o + 2, gate * (m.z + b.z));
            unsafeAtomicAdd(o + 3, gate * (m.w + b.w));
        }
        // "out" direction (flipped): gather dst, scatter to src
        {
            const float gate = sigmoidf_(g_out[d] + b_out_g[l]);
            const float4 m = *(const float4*)(Yo + (size_t)d * DOUT + c);
            const float4 b = *(const float4*)(b_out + (size_t)l * DOUT + c);
            float* o = out + (size_t)s * DOUT + c;
            unsafeAtomicAdd(o + 0, gate * (m.x + b.x));
            unsafeAtomicAdd(o + 1, gate * (m.y + b.y));
            unsafeAtomicAdd(o + 2, gate * (m.z + b.z));
            unsafeAtomicAdd(o + 3, gate * (m.w + b.w));
        }
    }
}

// ---------------------------------------------------------------------------
// K4: in-place ReLU on d_out.
// ---------------------------------------------------------------------------
__global__ __launch_bounds__(256)
void sgcn_relu_kernel(float* __restrict__ out, int n4)
{
    const int i = blockIdx.x * blockDim.x + threadIdx.x;
    if (i < n4) {
        float4 v = ((float4*)out)[i];
        v.x = fmaxf(v.x, 0.0f);
        v.y = fmaxf(v.y, 0.0f);
        v.z = fmaxf(v.z, 0.0f);
        v.w = fmaxf(v.w, 0.0f);
        ((float4*)out)[i] = v;
    }
}

extern "C" void kernel_launch(void* const* d_in, const int* in_sizes, int n_in,
                              void* d_out, int out_size, void* d_ws, size_t ws_size,
                              hipStream_t stream)
{
    const float* x        = (const float*)d_in[0];
    const int*   ei       = (const int*)  d_in[1];
    const int*   lab      = (const int*)  d_in[2];
    const float* W_loop   = (const float*)d_in[3];
    const float* b_loop   = (const float*)d_in[4];
    const float* W_loop_g = (const float*)d_in[5];
    const float* b_loop_g = (const float*)d_in[6];
    const float* W_in     = (const float*)d_in[7];
    const float* b_in     = (const float*)d_in[8];
    const float* W_in_g   = (const float*)d_in[9];
    const float* b_in_g   = (const float*)d_in[10];
    const float* W_out    = (const float*)d_in[11];
    const float* b_out    = (const float*)d_in[12];
    const float* W_out_g  = (const float*)d_in[13];
    const float* b_out_g  = (const float*)d_in[14];

    const int N = in_sizes[0] / DIN;
    const int E = in_sizes[2];

    float* out = (float*)d_out;
    float* Yi  = (float*)d_ws;                  // [N,128]
    float* Yo  = Yi + (size_t)N * DOUT;         // [N,128]
    float* gi  = Yo + (size_t)N * DOUT;         // [N]
    float* go  = gi + N;                        // [N]

    // K1: three GEMMs (Y_loop -> d_out, Y_in -> Yi, Y_out -> Yo)
    dim3 g1((N + 15) / 16, 1, 3);
    sgcn_gemm_kernel<<<g1, 256, 0, stream>>>(x, W_loop, W_in, W_out,
                                             out, Yi, Yo, N);

    // K2: gate dots + gated self-loop combine (one wave per node)
    int blocks2 = (N + 7) / 8;  // 8 waves per 256-thread block
    sgcn_gate_kernel<<<blocks2, 256, 0, stream>>>(x, W_loop_g, b_loop_g,
                                                  W_in_g, W_out_g, b_loop,
                                                  out, gi, go, N);

    // K3: edge gather/scatter (both directions)
    sgcn_edge_kernel<<<2048, 256, 0, stream>>>(Yi, Yo, gi, go, ei, lab,
                                               b_in, b_in_g, b_out, b_out_g,
                                               out, E);

    // K4: ReLU
    const int n4 = (N * DOUT) / 4;
    sgcn_relu_kernel<<<(n4 + 255) / 256, 256, 0, stream>>>(out, n4);
}